// MVN_elbo_autoMuV_32177894981903
// MI455X (gfx1250) — compile-verified
//
#include <hip/hip_runtime.h>
#include <hip/hip_bf16.h>

// ---------------------------------------------------------------------------
// GP posterior mean pipeline for MI455X (gfx1250, wave32, WMMA + TDM).
// Dominant cost: 16x Cholesky(2048) in fp32 -> V_WMMA_F32_16X16X4_F32 SYRK,
// blocked 64x64 per workgroup with Tensor-Data-Mover LDS staging.
// Fragment layouts per CDNA5 ISA 7.12.2:
//   A (16x4 f32): lane = M + 16*(K>=2), vgpr = K&1          -> v2f per lane
//   B (4x16 f32): lane = N + 16*(K>=2), vgpr = K&1          -> v2f per lane
//   C/D (16x16 f32): N = lane&15, M = vgpr + 8*(lane>>4)    -> v8f per lane
// ---------------------------------------------------------------------------

typedef __attribute__((ext_vector_type(2))) float v2f;
typedef __attribute__((ext_vector_type(8))) float v8f;
typedef __attribute__((ext_vector_type(4))) unsigned int tdm_g0_t;
typedef __attribute__((ext_vector_type(8))) int tdm_g1_t;
typedef __attribute__((ext_vector_type(4))) int tdm_g4_t;
typedef __attribute__((ext_vector_type(8))) int tdm_g8_t;

#define GN   2048   // N == N0 == M
#define GD   8
#define GKAP 16
#define GNB  64     // Cholesky panel width
#define LDP  65     // padded LDS row stride (64 + 1 DWORD TDM pad)
#define NUGGET 1e-6f

#define WMMA_F32(a, b, c) \
  __builtin_amdgcn_wmma_f32_16x16x4_f32(false, (a), false, (b), (short)0, (c), false, false)

// ---------------------------------------------------------------------------
// TDM: DMA one 64x64 f32 tile (row stride GN) from global into LDS with a
// 1-DWORD pad inserted every 64 DWORDs (-> stride-65 LDS layout).
// D# per ISA §8: group0 {count=1, lds_addr, global_addr[56:0], type=2},
// group1 {data_size=4B, pad_enable, pad_interval=64dw, pad_amount=1dw,
//         tensor_dim = tile_dim = 64x64, tensor_dim0_stride = GN}.
// This toolchain exposes the 6-arg builtin:
//   (u32x4 g0, i32x8 g1, i32x4 g2, i32x4 g3, i32x8 extra, i32 cpol)
#if __has_builtin(__builtin_amdgcn_tensor_load_to_lds)
#define HAVE_TDM 1
__device__ __forceinline__ void tdm_load_tile64(const float* base, int row0,
                                                int col0, unsigned lds_off) {
  unsigned long long ga =
      (unsigned long long)(uintptr_t)(base + (size_t)row0 * GN + col0);
  tdm_g0_t g0;
  g0[0] = 1u;                                            // count=1, user mode
  g0[1] = lds_off;                                       // lds_addr (bytes)
  g0[2] = (unsigned)(ga & 0xffffffffu);                  // global_addr[31:0]
  g0[3] = (unsigned)((ga >> 32) & 0x01ffffffu)           // global_addr[56:32]
          | (2u << 30);                                  // type=2 ("image")
  tdm_g1_t g1;
  g1[0] = (0x2 << 16)                                    // data_size = 4B
          | (1 << 20)                                    // pad_enable
          | (5 << 22)                                    // pad_interval: 64 dw
          | (0 << 25);                                   // pad_amount: 1 dw
  g1[1] = (GNB & 0xffff) << 16;                          // tensor_dim0[15:0]
  g1[2] = (GNB & 0xffff) << 16;                          // tensor_dim1[15:0]
  g1[3] = (GNB & 0xffff) << 16;                          // tile_dim0 = 64
  g1[4] = GNB;                                           // tile_dim1 = 64
  g1[5] = GN;                                            // dim0_stride[31:0]
  g1[6] = 0;
  g1[7] = 0;
  tdm_g4_t z4 = {0, 0, 0, 0};
  tdm_g8_t z8 = {0, 0, 0, 0, 0, 0, 0, 0};
  __builtin_amdgcn_tensor_load_to_lds(g0, g1, z4, z4, z8, 0);
}
#else
#define HAVE_TDM 0
#endif

// ---------------------------------------------------------------------------
// K1: xs = theta * exp(-lmb), x0s = theta0 * exp(-lmb), row squared norms.
__global__ __launch_bounds__(256)
void scale_rows(const float* __restrict__ theta, const float* __restrict__ theta0,
                const float* __restrict__ Lmb_k,
                float* __restrict__ xs, float* __restrict__ x0s,
                float* __restrict__ s, float* __restrict__ s0) {
  int r = blockIdx.x * 256 + threadIdx.x;        // 0..2*GN-1
  float inv[GD];
#pragma unroll
  for (int d = 0; d < GD; ++d) inv[d] = __expf(-Lmb_k[d]);
  const float* src; float* dst; float* sv; int row;
  if (r < GN) { src = theta;  dst = xs;  sv = s;  row = r; }
  else        { src = theta0; dst = x0s; sv = s0; row = r - GN; }
  float acc = 0.f;
#pragma unroll
  for (int d = 0; d < GD; ++d) {
    float v = src[row * GD + d] * inv[d];
    dst[row * GD + d] = v;
    acc += v * v;
  }
  sv[row] = acc;
}

// ---------------------------------------------------------------------------
// K2: R[i,j] = scale*exp(-0.5*max(s[i]+s[j]-2*<xs_i,xs_j>,0)) + nugget*delta.
__global__ __launch_bounds__(32)
void build_R(const float* __restrict__ xs, const float* __restrict__ s,
             const float* __restrict__ Lmb_k, float* __restrict__ R) {
  const int lane = threadIdx.x;
  const int m  = lane & 15;
  const int kh = lane >> 4;
  const int row0 = blockIdx.y * 16, col0 = blockIdx.x * 16;
  v8f acc = {};
#pragma unroll
  for (int k0 = 0; k0 < GD; k0 += 4) {
    v2f a, b;
    a.x = xs[(row0 + m) * GD + k0 + 2 * kh + 0];
    a.y = xs[(row0 + m) * GD + k0 + 2 * kh + 1];
    b.x = xs[(col0 + m) * GD + k0 + 2 * kh + 0];   // B[k][n] = xs[n][k]
    b.y = xs[(col0 + m) * GD + k0 + 2 * kh + 1];
    acc = WMMA_F32(a, b, acc);
  }
  float scale = __expf(Lmb_k[GD]);
#pragma unroll
  for (int i = 0; i < 8; ++i) {
    int gi = row0 + i + 8 * kh;
    int gj = col0 + m;
    float d2 = fmaxf(s[gi] + s[gj] - 2.0f * acc[i], 0.0f);
    float rv = scale * __expf(-0.5f * d2);
    if (gi == gj) rv += NUGGET;
    R[gi * GN + gj] = rv;
  }
}

// ---------------------------------------------------------------------------
// K3: unblocked Cholesky of the 64x64 diagonal block (lower), in LDS.
__global__ __launch_bounds__(64)
void potf2(float* __restrict__ R, int j1) {
  __shared__ float Ls[GNB * GNB];
  int tid = threadIdx.x;
  for (int idx = tid; idx < GNB * GNB; idx += 64)
    Ls[idx] = R[(j1 + idx / GNB) * GN + j1 + (idx & (GNB - 1))];
  __syncthreads();
  for (int j = 0; j < GNB; ++j) {
    if (tid == 0) Ls[j * GNB + j] = sqrtf(Ls[j * GNB + j]);
    __syncthreads();
    float dj = Ls[j * GNB + j];
    if (tid > j) Ls[tid * GNB + j] /= dj;
    __syncthreads();
    if (tid > j) {
      float lij = Ls[tid * GNB + j];
      for (int c = j + 1; c <= tid; ++c)
        Ls[tid * GNB + c] -= lij * Ls[c * GNB + j];
    }
    __syncthreads();
  }
  for (int idx = tid; idx < GNB * GNB; idx += 64)
    R[(j1 + idx / GNB) * GN + j1 + (idx & (GNB - 1))] = Ls[idx];
}

// ---------------------------------------------------------------------------
// K4: TRSM  L21 = A21 * L11^{-T}: one thread per row, forward substitution
// against the LDS-resident 64x64 lower-triangular diagonal block.
__global__ __launch_bounds__(64)
void trsm(float* __restrict__ R, int j1) {
  __shared__ float Ls[GNB * GNB];
  __shared__ float rs[64 * LDP];
  int tid = threadIdx.x;
  for (int idx = tid; idx < GNB * GNB; idx += 64)
    Ls[idx] = R[(j1 + idx / GNB) * GN + j1 + (idx & (GNB - 1))];
  __syncthreads();
  int row = j1 + GNB + blockIdx.x * 64 + tid;
  for (int j = 0; j < GNB; ++j) rs[tid * LDP + j] = R[row * GN + j1 + j];
  for (int j = 0; j < GNB; ++j) {
    float a = rs[tid * LDP + j];
    for (int i = 0; i < j; ++i) a -= Ls[j * GNB + i] * rs[tid * LDP + i];
    rs[tid * LDP + j] = a / Ls[j * GNB + j];
  }
  for (int j = 0; j < GNB; ++j) R[row * GN + j1 + j] = rs[tid * LDP + j];
}

// ---------------------------------------------------------------------------
// K5: SYRK trailing update, 64x64 tile per 128-thread workgroup.
// Panels staged into LDS by the Tensor Data Mover (wave 0 issues two D#,
// s_wait_tensorcnt 0, workgroup barrier). Each of the 4 waves owns a 32x32
// quadrant = 2x2 accumulators; K=64 -> 64 wmma_f32_16x16x4 per wave
// with 2x fragment reuse. LDS rows stride-65 (TDM pad) -> conflict-free.
__global__ __launch_bounds__(128)
void syrk(float* __restrict__ R, int j1) {
  int tj = blockIdx.x, ti = blockIdx.y;
  if (ti < tj) return;                       // lower (incl. diagonal) tiles
  __shared__ float As[GNB * LDP];
  __shared__ float Bs[GNB * LDP];
  const int t0 = j1 + GNB;
  const int tid = threadIdx.x;
  const int waveId = tid >> 5;
  const int lane = tid & 31;

#if HAVE_TDM
  if (waveId == 0) {
    tdm_load_tile64(R, t0 + ti * GNB, j1, (unsigned)(uintptr_t)As);
    tdm_load_tile64(R, t0 + tj * GNB, j1, (unsigned)(uintptr_t)Bs);
    __builtin_amdgcn_s_wait_tensorcnt(0);
  }
#else
  for (int idx = tid; idx < GNB * GNB; idx += 128) {
    int r = idx >> 6, c = idx & (GNB - 1);
    As[r * LDP + c] = R[(t0 + ti * GNB + r) * GN + j1 + c];
    Bs[r * LDP + c] = R[(t0 + tj * GNB + r) * GN + j1 + c];
  }
#endif
  __syncthreads();

  const int wi = waveId >> 1, wj = waveId & 1;   // 2x2 waves over 64x64
  const int m  = lane & 15;
  const int kh = lane >> 4;
  const int ar0 = wi * 32;                       // local A-row base
  const int bc0 = wj * 32;                       // local B-col base
  v8f acc00 = {}, acc01 = {}, acc10 = {}, acc11 = {};
#pragma unroll
  for (int k0 = 0; k0 < GNB; k0 += 4) {
    v2f a0, a1, b0, b1;
    a0.x = As[(ar0 + m)      * LDP + k0 + 2 * kh + 0];
    a0.y = As[(ar0 + m)      * LDP + k0 + 2 * kh + 1];
    a1.x = As[(ar0 + 16 + m) * LDP + k0 + 2 * kh + 0];
    a1.y = As[(ar0 + 16 + m) * LDP + k0 + 2 * kh + 1];
    b0.x = Bs[(bc0 + m)      * LDP + k0 + 2 * kh + 0];
    b0.y = Bs[(bc0 + m)      * LDP + k0 + 2 * kh + 1];
    b1.x = Bs[(bc0 + 16 + m) * LDP + k0 + 2 * kh + 0];
    b1.y = Bs[(bc0 + 16 + m) * LDP + k0 + 2 * kh + 1];
    acc00 = WMMA_F32(a0, b0, acc00);
    acc01 = WMMA_F32(a0, b1, acc01);
    acc10 = WMMA_F32(a1, b0, acc10);
    acc11 = WMMA_F32(a1, b1, acc11);
  }
  // writeback: T -= L21 * L21^T  (f32 WMMA has no A/B negation per ISA)
  const int gi0 = t0 + ti * GNB + ar0;
  const int gj0 = t0 + tj * GNB + bc0;
#pragma unroll
  for (int i = 0; i < 8; ++i) {
    int r0 = gi0 + i + 8 * kh;
    int r1 = r0 + 16;
    R[r0 * GN + gj0 + m]      -= acc00[i];
    R[r0 * GN + gj0 + 16 + m] -= acc01[i];
    R[r1 * GN + gj0 + m]      -= acc10[i];
    R[r1 * GN + gj0 + 16 + m] -= acc11[i];
  }
}

// ---------------------------------------------------------------------------
// K6: rg = L^{-T} L^{-1} Mu[k]  (column-sweep fwd + bwd substitution in LDS).
__global__ __launch_bounds__(1024)
void solve(const float* __restrict__ R, const float* __restrict__ Mu,
           int k, float* __restrict__ rg) {
  __shared__ float ys[GN];
  int tid = threadIdx.x;
  for (int i = tid; i < GN; i += 1024) ys[i] = Mu[k * GN + i];
  __syncthreads();
  for (int j = 0; j < GN; ++j) {            // forward: L y = g
    if (tid == 0) ys[j] = ys[j] / R[j * GN + j];
    __syncthreads();
    float yj = ys[j];
    for (int i = j + 1 + tid; i < GN; i += 1024) ys[i] -= R[i * GN + j] * yj;
    __syncthreads();
  }
  for (int j = GN - 1; j >= 0; --j) {       // backward: L^T x = y
    if (tid == 0) ys[j] = ys[j] / R[j * GN + j];
    __syncthreads();
    float xj = ys[j];
    for (int i = tid; i < j; i += 1024) ys[i] -= R[j * GN + i] * xj;
    __syncthreads();
  }
  for (int i = tid; i < GN; i += 1024) rg[i] = ys[i];
}

// ---------------------------------------------------------------------------
// K7: ghat[k][i0] = scale * sum_n exp(-0.5*d2(i0,n)) * rg[n]   (fused, no R_no)
__global__ __launch_bounds__(256)
void ghat_gemv(const float* __restrict__ xs, const float* __restrict__ x0s,
               const float* __restrict__ s, const float* __restrict__ s0,
               const float* __restrict__ rg, const float* __restrict__ Lmb_k,
               float* __restrict__ ghat_k) {
  __shared__ float xt[256 * GD];
  __shared__ float st[256];
  __shared__ float rt[256];
  int tid = threadIdx.x;
  int i0 = blockIdx.x * 256 + tid;
  float x0r[GD];
#pragma unroll
  for (int d = 0; d < GD; ++d) x0r[d] = x0s[i0 * GD + d];
  float s0i = s0[i0];
  float acc = 0.f;
  for (int nb = 0; nb < GN; nb += 256) {
    __syncthreads();
#pragma unroll
    for (int d = 0; d < GD; ++d) xt[tid * GD + d] = xs[(nb + tid) * GD + d];
    st[tid] = s[nb + tid];
    rt[tid] = rg[nb + tid];
    __syncthreads();
    for (int j = 0; j < 256; ++j) {
      float dot = 0.f;
#pragma unroll
      for (int d = 0; d < GD; ++d) dot += x0r[d] * xt[j * GD + d];
      float d2 = fmaxf(s0i + st[j] - 2.0f * dot, 0.0f);
      acc += __expf(-0.5f * d2) * rt[j];
    }
  }
  ghat_k[i0] = __expf(Lmb_k[GD]) * acc;
}

// ---------------------------------------------------------------------------
// K8: fhat = psi + Phi @ ghat   (K=16 -> 4x wmma per 16x16 tile)
__global__ __launch_bounds__(32)
void final_gemm(const float* __restrict__ Phi, const float* __restrict__ psi,
                const float* __restrict__ ghat, float* __restrict__ out) {
  int lane = threadIdx.x;
  int m  = lane & 15;
  int kh = lane >> 4;
  int row0 = blockIdx.y * 16, col0 = blockIdx.x * 16;
  v8f acc = {};
#pragma unroll
  for (int k0 = 0; k0 < GKAP; k0 += 4) {
    v2f a, b;
    a.x = Phi[(row0 + m) * GKAP + k0 + 2 * kh + 0];
    a.y = Phi[(row0 + m) * GKAP + k0 + 2 * kh + 1];
    b.x = ghat[(k0 + 2 * kh + 0) * GN + col0 + m];
    b.y = ghat[(k0 + 2 * kh + 1) * GN + col0 + m];
    acc = WMMA_F32(a, b, acc);
  }
#pragma unroll
  for (int i = 0; i < 8; ++i) {
    int gm = row0 + i + 8 * kh;
    out[gm * GN + col0 + m] = psi[gm] + acc[i];
  }
}

// ---------------------------------------------------------------------------
extern "C" void kernel_launch(void* const* d_in, const int* in_sizes, int n_in,
                              void* d_out, int out_size, void* d_ws, size_t ws_size,
                              hipStream_t stream) {
  const float* theta0 = (const float*)d_in[0];   // (2048, 8)
  const float* theta  = (const float*)d_in[1];   // (2048, 8)
  const float* Lmb    = (const float*)d_in[2];   // (16, 9)
  const float* Mu     = (const float*)d_in[3];   // (16, 2048)
  const float* psi    = (const float*)d_in[4];   // (2048, 1)
  const float* Phi    = (const float*)d_in[5];   // (2048, 16)
  float* out = (float*)d_out;                    // (2048, 2048)

  // workspace carve-up (~17 MB; R reused across k so L2-resident)
  float* ws   = (float*)d_ws;
  float* R    = ws;                         // GN*GN
  float* xs   = R + (size_t)GN * GN;        // GN*GD
  float* x0s  = xs + GN * GD;               // GN*GD
  float* s    = x0s + GN * GD;              // GN
  float* s0   = s + GN;                     // GN
  float* rg   = s0 + GN;                    // GN
  float* ghat = rg + GN;                    // GKAP*GN

  for (int k = 0; k < GKAP; ++k) {
    const float* Lmbk = Lmb + k * (GD + 1);
    scale_rows<<<(2 * GN) / 256, 256, 0, stream>>>(theta, theta0, Lmbk,
                                                   xs, x0s, s, s0);
    build_R<<<dim3(GN / 16, GN / 16), 32, 0, stream>>>(xs, s, Lmbk, R);

    for (int jb = 0; jb < GN / GNB; ++jb) {
      int j1 = jb * GNB;
      potf2<<<1, 64, 0, stream>>>(R, j1);
      int rows = GN - (j1 + GNB);
      if (rows > 0) {
        trsm<<<rows / 64, 64, 0, stream>>>(R, j1);
        int nt64 = rows / GNB;
        syrk<<<dim3(nt64, nt64), 128, 0, stream>>>(R, j1);
      }
    }

    solve<<<1, 1024, 0, stream>>>(R, Mu, k, rg);
    ghat_gemv<<<GN / 256, 256, 0, stream>>>(xs, x0s, s, s0, rg, Lmbk,
                                            ghat + (size_t)k * GN);
  }

  final_gemm<<<dim3(GN / 16, GN / 16), 32, 0, stream>>>(Phi, psi, ghat, out);
}